// PyramidSAK_61735859913296
// MI455X (gfx1250) — compile-verified
//
#include <hip/hip_runtime.h>

#define BATCH 16
#define NPTS  8192
#define NSAMP 512
#define KMAX  24

typedef float v2f __attribute__((ext_vector_type(2)));
typedef float v8f __attribute__((ext_vector_type(8)));

// ---------------- Farthest Point Sampling: one workgroup per batch ----------------
// Serial 512-round argmax; each thread owns 32 points in registers.
__global__ __launch_bounds__(256) void fps_kernel(const float* __restrict__ l0_xyz,
                                                  float* __restrict__ newxyz,
                                                  float* __restrict__ out) {
  const int b = blockIdx.x;
  const int tid = threadIdx.x;
  const float* X = l0_xyz + (size_t)b * 3 * NPTS;
  const int base = tid * 32;
  float px[32], py[32], pz[32], dist[32];
#pragma unroll
  for (int k = 0; k < 32; ++k) {
    px[k] = X[base + k];
    py[k] = X[NPTS + base + k];
    pz[k] = X[2 * NPTS + base + k];
    dist[k] = 1e10f;
  }
  __shared__ float sd[256];
  __shared__ int   si[256];
  int far = 0;
  for (int it = 0; it < NSAMP; ++it) {
    const float cx = X[far], cy = X[NPTS + far], cz = X[2 * NPTS + far];
    if (tid == 0) {
      float* nx = newxyz + ((size_t)b * NSAMP + it) * 3;
      nx[0] = cx; nx[1] = cy; nx[2] = cz;
      out[(size_t)b * 3 * NSAMP + 0 * NSAMP + it] = cx;   // keypoints [B,3,S]
      out[(size_t)b * 3 * NSAMP + 1 * NSAMP + it] = cy;
      out[(size_t)b * 3 * NSAMP + 2 * NSAMP + it] = cz;
    }
    float bestd = -1.0f; int besti = base;
#pragma unroll
    for (int k = 0; k < 32; ++k) {
      const float dx = px[k] - cx, dy = py[k] - cy, dz = pz[k] - cz;
      const float d = dx * dx + dy * dy + dz * dz;
      const float dm = fminf(dist[k], d);
      dist[k] = dm;
      if (dm > bestd) { bestd = dm; besti = base + k; }  // strict > : lowest index on tie
    }
    sd[tid] = bestd; si[tid] = besti;
    __syncthreads();
    for (int off = 128; off > 0; off >>= 1) {
      if (tid < off) {
        if (sd[tid + off] > sd[tid]) { sd[tid] = sd[tid + off]; si[tid] = si[tid + off]; }
      }
      __syncthreads();
    }
    far = si[0];
    __syncthreads();
  }
}

// ---------------- top-24 nearest neighbors, one thread per (b,s) ----------------
__global__ __launch_bounds__(256) void knn_kernel(const float* __restrict__ l0_xyz,
                                                  const float* __restrict__ newxyz,
                                                  int* __restrict__ knn) {
  const int gid = blockIdx.x * 256 + threadIdx.x;   // [0, B*S)
  const int b = gid >> 9;
  const float* X = l0_xyz + (size_t)b * 3 * NPTS;
  const float cx = newxyz[gid * 3 + 0];
  const float cy = newxyz[gid * 3 + 1];
  const float cz = newxyz[gid * 3 + 2];
  float bd[KMAX]; int bi[KMAX];
#pragma unroll
  for (int j = 0; j < KMAX; ++j) { bd[j] = 3.4e38f; bi[j] = 0; }
  for (int n = 0; n < NPTS; ++n) {
    __builtin_prefetch(&X[n + 128], 0, 1);
    const float dx = X[n] - cx, dy = X[NPTS + n] - cy, dz = X[2 * NPTS + n] - cz;
    const float d = dx * dx + dy * dy + dz * dz;
    if (d < bd[KMAX - 1]) {            // strict < : lower index wins ties (top_k order)
      int j = KMAX - 1;
      while (j > 0 && bd[j - 1] > d) { bd[j] = bd[j - 1]; bi[j] = bi[j - 1]; --j; }
      bd[j] = d; bi[j] = n;
    }
  }
#pragma unroll
  for (int j = 0; j < KMAX; ++j) knn[(size_t)gid * KMAX + j] = bi[j];
}

// ---------------- fused grouped MLP (layer1 VALU -> LDS, layer2 WMMA f32, maxpool) ----
// grid.x : 8 groups (waves) per block over B*S groups; grid.y : head 0..3
__global__ __launch_bounds__(256) void mlp_kernel(
    const float* __restrict__ newxyz, const int* __restrict__ knn,
    const float* __restrict__ l0_xyz, const float* __restrict__ l0_pts,
    const float* __restrict__ saW1, const float* __restrict__ sab1,
    const float* __restrict__ saW2, const float* __restrict__ sab2,
    const float* __restrict__ msW1, const float* __restrict__ msb1,
    const float* __restrict__ msW2, const float* __restrict__ msb2,
    float* __restrict__ out) {
  __shared__ float hbuf[8][16 * 65];          // 16x64 activation tile per wave (pad 65)
  const int wave = threadIdx.x >> 5;
  const int lane = threadIdx.x & 31;
  const int head = blockIdx.y;                // block-uniform
  const int g = blockIdx.x * 8 + wave;
  const int b = g >> 9;
  const int s = g & 511;

  int K, cin;
  const float *W1, *b1, *W2, *b2;
  if (head == 0) { K = 16; cin = 6; W1 = saW1; b1 = sab1; W2 = saW2; b2 = sab2; }
  else {
    const int i = head - 1;
    K = (i == 0) ? 8 : ((i == 1) ? 16 : 24);
    cin = 3;
    W1 = msW1 + i * 64 * 3; b1 = msb1 + i * 64;
    W2 = msW2 + i * 128 * 64; b2 = msb2 + i * 128;
  }
  const float cx = newxyz[g * 3 + 0], cy = newxyz[g * 3 + 1], cz = newxyz[g * 3 + 2];
  const float* X = l0_xyz + (size_t)b * 3 * NPTS;
  const float* P = l0_pts + (size_t)b * 3 * NPTS;
  const int* kn = knn + (size_t)g * KMAX;

  float red[8];
#pragma unroll
  for (int nt = 0; nt < 8; ++nt) red[nt] = -3.4e38f;

  const int ntiles = (K + 15) >> 4;           // 1 or 2 (uniform per block)
  float* hw = &hbuf[wave][0];
  for (int mt = 0; mt < ntiles; ++mt) {
    const int vrows = (K - mt * 16 < 16) ? (K - mt * 16) : 16;   // 8 or 16
    // ---- layer 1: rel-coord (+point feat) MLP 3/6 -> 64, ReLU, to LDS ----
    for (int t = 0; t < 32; ++t) {
      const int row = t >> 1;
      const int ch = ((t & 1) << 5) | lane;
      float hv = 0.0f;
      if (row < vrows) {
        const int n = kn[mt * 16 + row];
        const float rx = X[n] - cx, ry = X[NPTS + n] - cy, rz = X[2 * NPTS + n] - cz;
        const float* w = W1 + ch * cin;
        float acc = b1[ch] + rx * w[0] + ry * w[1] + rz * w[2];
        if (cin == 6)
          acc += P[n] * w[3] + P[NPTS + n] * w[4] + P[2 * NPTS + n] * w[5];
        hv = fmaxf(acc, 0.0f);
      }
      hw[row * 65 + ch] = hv;                 // zero-pad invalid rows
    }
    __syncthreads();
    // ---- layer 2: [16,64] x [64,128] via V_WMMA_F32_16X16X4_F32 ----
    const int rowA = lane & 15;
    const int koff = (lane < 16) ? 0 : 2;     // A: lanes<16 hold K=0,1; lanes>=16 K=2,3
    const float* hrow = hw + rowA * 65;
    for (int nt = 0; nt < 8; ++nt) {
      const int n = nt * 16 + (lane & 15);    // output channel within 128
      const float bias = b2[n];
      v8f acc;
#pragma unroll
      for (int r = 0; r < 8; ++r) acc[r] = bias;   // bias depends only on N
      const float* wcol = W2 + (size_t)n * 64;     // W2[o][p], p contiguous
#pragma unroll
      for (int kk = 0; kk < 64; kk += 4) {
        v2f a, bb;
        a.x  = hrow[kk + koff];
        a.y  = hrow[kk + koff + 1];
        bb.x = wcol[kk + koff];               // B: K rows striped across lane halves
        bb.y = wcol[kk + koff + 1];
        acc = __builtin_amdgcn_wmma_f32_16x16x4_f32(false, a, false, bb,
                                                    (short)0, acc, false, false);
      }
      // max over M: VGPR r holds M=r (lanes<16) / M=r+8 (lanes>=16)
      float m = acc[0];
#pragma unroll
      for (int r = 1; r < 8; ++r) m = fmaxf(m, acc[r]);
      const float o = __shfl_xor(m, 16, 32);
      if (vrows == 16) m = fmaxf(m, o);       // full tile: combine halves
      else             m = (lane < 16) ? m : o; // 8 valid rows: lanes<16 half only
      red[nt] = fmaxf(red[nt], m);
    }
    __syncthreads();
  }
  if (lane < 16) {
#pragma unroll
    for (int nt = 0; nt < 8; ++nt) {
      const int ch = head * 128 + nt * 16 + lane;
      out[24576 + (size_t)b * 512 * NSAMP + (size_t)ch * NSAMP + s] = red[nt];
    }
  }
}

extern "C" void kernel_launch(void* const* d_in, const int* in_sizes, int n_in,
                              void* d_out, int out_size, void* d_ws, size_t ws_size,
                              hipStream_t stream) {
  (void)in_sizes; (void)n_in; (void)out_size; (void)ws_size;
  const float* l0_xyz = (const float*)d_in[0];
  const float* l0_pts = (const float*)d_in[1];
  const float* saW1   = (const float*)d_in[2];
  const float* sab1   = (const float*)d_in[3];
  const float* saW2   = (const float*)d_in[4];
  const float* sab2   = (const float*)d_in[5];
  const float* msW1   = (const float*)d_in[6];
  const float* msb1   = (const float*)d_in[7];
  const float* msW2   = (const float*)d_in[8];
  const float* msb2   = (const float*)d_in[9];
  float* out = (float*)d_out;

  float* newxyz = (float*)d_ws;                                   // [B,S,3]
  int*   knn    = (int*)((char*)d_ws + (size_t)BATCH * NSAMP * 3 * sizeof(float)); // [B,S,24]

  fps_kernel<<<dim3(BATCH), dim3(256), 0, stream>>>(l0_xyz, newxyz, out);
  knn_kernel<<<dim3((BATCH * NSAMP) / 256), dim3(256), 0, stream>>>(l0_xyz, newxyz, knn);
  mlp_kernel<<<dim3((BATCH * NSAMP) / 8, 4), dim3(256), 0, stream>>>(
      newxyz, knn, l0_xyz, l0_pts, saW1, sab1, saW2, sab2,
      msW1, msb1, msW2, msb2, out);
}